// EGNN_10634339025285
// MI455X (gfx1250) — compile-verified
//
#include <hip/hip_runtime.h>
#include <math.h>

// ---------------- problem constants ----------------
#define NN 100000      // nodes
#define NE 3200000     // edges
#define NG 1000        // graphs
#define HD 16          // hidden dim

typedef __attribute__((ext_vector_type(2))) float v2f;
typedef __attribute__((ext_vector_type(8))) float v8f;

// ---------------- utility kernels ----------------
__global__ void zero_f32(float* __restrict__ p, int n) {
    int i = blockIdx.x * blockDim.x + threadIdx.x;
    if (i < n) p[i] = 0.0f;
}

// in-degree over target column (self-loop added analytically in dinv_kernel)
__global__ void deg_kernel(const int* __restrict__ col, float* __restrict__ deg) {
    int e = blockIdx.x * blockDim.x + threadIdx.x;
    if (e < NE) atomicAdd(&deg[col[e]], 1.0f);
}

__global__ void dinv_kernel(const float* __restrict__ deg, float* __restrict__ dinv) {
    int i = blockIdx.x * blockDim.x + threadIdx.x;
    if (i < NN) {
        float d = deg[i] + 1.0f;        // +1 for self loop; always > 0
        dinv[i] = rsqrtf(d);
    }
}

__global__ void norm_kernel(const int* __restrict__ row, const int* __restrict__ col,
                            const float* __restrict__ dinv, float* __restrict__ nrm) {
    int e = blockIdx.x * blockDim.x + threadIdx.x;
    if (e < NE) nrm[e] = dinv[row[e]] * dinv[col[e]];
}

// ---- fused WMMA node transform + accumulator init ----
// h   = in @ W                      (written to h_out, feeds the edge gather)
// acc = bias + dinv^2 * h           (self-loop + bias term, written to acc_out)
//
// acc_out may alias `in`: each 16-row tile is read by exactly one wave (A fragment)
// and written by exactly that wave, and the wave's loads complete (s_wait_loadcnt)
// before its stores issue -> safe in-place overwrite, no cross-wave sharing.
//
// One wave per 16-row tile, V_WMMA_F32_16X16X4_F32 chained over K.
// Layouts (ISA 7.12.2, 32-bit):
//   A 16x4 : lane l -> M = l&15, VGPR v -> K = 2*(l>>4) + v
//   B 4x16 : lane l -> N = l&15, VGPR v -> K = 2*(l>>4) + v
//   C 16x16: lane l, VGPR r -> M = r + 8*(l>>4), N = l&15
// blockDim=64 (2 waves), grid = (NN/16)/2 = 3125 -> no partial waves, EXEC all-1s.
template <int KDIM>
__global__ __launch_bounds__(64)
void transform_fused(const float* __restrict__ in, const float* __restrict__ W,
                     const float* __restrict__ bias, const float* __restrict__ dinv,
                     float* __restrict__ h_out, float* __restrict__ acc_out) {
    int wave = (blockIdx.x * blockDim.x + threadIdx.x) >> 5;   // global tile id
    int lane = threadIdx.x & 31;
    int m    = lane & 15;
    int kg   = lane >> 4;          // 0 or 1
    int base = wave * 16;          // first row of this tile

    float bcol = bias[m];          // bias for this lane's output column

    v8f c = {};
#pragma unroll
    for (int k = 0; k < KDIM / 4; ++k) {
        v2f a, b;
        // A fragment: contiguous pair of input elements -> one b64 load
        const float* ap = in + (size_t)(base + m) * KDIM + 4 * k + 2 * kg;
        a[0] = ap[0];
        a[1] = ap[1];
        // B fragment: rows of W
        int k0 = 4 * k + 2 * kg;
        b[0] = W[(k0 + 0) * HD + m];
        b[1] = W[(k0 + 1) * HD + m];
        c = __builtin_amdgcn_wmma_f32_16x16x4_f32(
                /*neg_a=*/false, a, /*neg_b=*/false, b,
                /*c_mod=*/(short)0, c, /*reuse_a=*/false, /*reuse_b=*/false);
    }

#pragma unroll
    for (int r = 0; r < 8; ++r) {
        int rowi = base + r + 8 * kg;
        float hv = c[r];
        float di = dinv[rowi];
        h_out[(size_t)rowi * HD + m]   = hv;
        acc_out[(size_t)rowi * HD + m] = bcol + di * di * hv;   // self-loop + bias
    }
}

// edge scatter: acc[col] += norm * h[row]; 4 threads per edge, float4 chunk each.
__global__ void scatter_kernel(const int* __restrict__ row, const int* __restrict__ col,
                               const float* __restrict__ nrm, const float* __restrict__ h,
                               float* __restrict__ out) {
    long long t = (long long)blockIdx.x * blockDim.x + threadIdx.x;
    if (t >= (long long)NE * 4) return;
    int e = (int)(t >> 2);
    int q = (int)(t & 3);
    int r = row[e], cc = col[e];
    float w = nrm[e];
    const float4* hp = (const float4*)(h + (size_t)r * HD + q * 4);
    float4 mv = *hp;
    float* op = out + (size_t)cc * HD + q * 4;
    atomicAdd(op + 0, mv.x * w);   // non-returning -> global_atomic_add_f32
    atomicAdd(op + 1, mv.y * w);
    atomicAdd(op + 2, mv.z * w);
    atomicAdd(op + 3, mv.w * w);
}

// mean-pool accumulation per graph
__global__ void pool_kernel(const float* __restrict__ h, const int* __restrict__ batch,
                            float* __restrict__ sums, float* __restrict__ cnt) {
    int i = blockIdx.x * blockDim.x + threadIdx.x;
    if (i < NN * HD) {
        int v = i >> 4, cidx = i & 15;
        int g = batch[v];
        atomicAdd(&sums[g * HD + cidx], h[i]);
        if (cidx == 0) atomicAdd(&cnt[g], 1.0f);
    }
}

// logits = pooled @ lin_w + lin_b ; sigmoid
__global__ void head_kernel(const float* __restrict__ sums, const float* __restrict__ cnt,
                            const float* __restrict__ lw, const float* __restrict__ lb,
                            float* __restrict__ out) {
    int g = blockIdx.x * blockDim.x + threadIdx.x;
    if (g < NG) {
        float invc = 1.0f / fmaxf(cnt[g], 1.0f);
        float acc = lb[0];
#pragma unroll
        for (int c = 0; c < HD; ++c) acc += sums[g * HD + c] * invc * lw[c];
        out[g] = 1.0f / (1.0f + expf(-acc));
    }
}

// ---------------- launcher ----------------
extern "C" void kernel_launch(void* const* d_in, const int* in_sizes, int n_in,
                              void* d_out, int out_size, void* d_ws, size_t ws_size,
                              hipStream_t stream) {
    const float* x     = (const float*)d_in[0];
    const int*   edges = (const int*)d_in[1];   // [2, NE]: first NE = row(src), next NE = col(dst)
    const int*   batch = (const int*)d_in[2];
    const float* W1 = (const float*)d_in[3];
    const float* b1 = (const float*)d_in[4];
    const float* W2 = (const float*)d_in[5];
    const float* b2 = (const float*)d_in[6];
    const float* W3 = (const float*)d_in[7];
    const float* b3 = (const float*)d_in[8];
    const float* lw = (const float*)d_in[9];
    const float* lb = (const float*)d_in[10];
    float* out = (float*)d_out;

    const int* row = edges;
    const int* col = edges + NE;

    // workspace carve-out (floats, 256B-aligned chunks)
    float* ws = (float*)d_ws;
    size_t off = 0;
    auto carve = [&](size_t n) { float* p = ws + off; off += (n + 63) & ~(size_t)63; return p; };
    float* deg  = carve(NN);
    float* dinv = carve(NN);
    float* nrm  = carve(NE);
    float* hA   = carve((size_t)NN * HD);   // transformed features (gather source)
    float* hB   = carve((size_t)NN * HD);   // accumulator / layer output
    float* sums = carve((size_t)NG * HD);   // NG*HD = 16000, multiple of 64 -> cnt contiguous
    float* cnt  = carve(NG);
    (void)ws_size;

    const int B = 256;
    const int gN  = (NN + B - 1) / B;        // 391
    const int gE  = NE / B;                  // 12500 (exact)
    const int gNH = (NN * HD) / B;           // 6250  (exact)
    const int gSc = (NE * 4) / B;            // 50000 (exact)
    const int gT  = (NN / 16) / 2;           // 3125  (exact)

    // --- normalization precompute ---
    zero_f32<<<gN, B, 0, stream>>>(deg, NN);
    deg_kernel<<<gE, B, 0, stream>>>(col, deg);
    dinv_kernel<<<gN, B, 0, stream>>>(deg, dinv);
    norm_kernel<<<gE, B, 0, stream>>>(row, col, dinv, nrm);

    // --- layer 1: x(NNx4) @ W1 ; acc = b1 + dinv^2*h ; scatter edges ---
    transform_fused<4><<<gT, 64, 0, stream>>>(x, W1, b1, dinv, hA, hB);
    scatter_kernel<<<gSc, B, 0, stream>>>(row, col, nrm, hA, hB);

    // --- layer 2: hB @ W2 (acc overwrites hB in place) ---
    transform_fused<16><<<gT, 64, 0, stream>>>(hB, W2, b2, dinv, hA, hB);
    scatter_kernel<<<gSc, B, 0, stream>>>(row, col, nrm, hA, hB);

    // --- layer 3: hB @ W3 ---
    transform_fused<16><<<gT, 64, 0, stream>>>(hB, W3, b3, dinv, hA, hB);
    scatter_kernel<<<gSc, B, 0, stream>>>(row, col, nrm, hA, hB);

    // --- global mean pool + linear head + sigmoid ---
    zero_f32<<<(NG * HD + NG + B - 1) / B, B, 0, stream>>>(sums, NG * HD + NG);
    pool_kernel<<<gNH, B, 0, stream>>>(hB, batch, sums, cnt);
    head_kernel<<<(NG + B - 1) / B, B, 0, stream>>>(sums, cnt, lw, lb, out);
}